// OLCNN_21208548507914
// MI455X (gfx1250) — compile-verified
//
#include <hip/hip_runtime.h>

typedef float v2f __attribute__((ext_vector_type(2)));
typedef float v8f __attribute__((ext_vector_type(8)));

#define WAVES_PER_BLOCK 4
#define SAMPLES_PER_GROUP 16

__device__ __forceinline__ float sigmoidf_fast(float x) {
    return 1.0f / (1.0f + __expf(-x));
}

__global__ __launch_bounds__(128) void olcnn_kernel(
    const float* __restrict__ x,    // (B, 81)
    const float* __restrict__ w1,   // (16, 9)
    const float* __restrict__ b1,   // (16)
    const float* __restrict__ w2,   // (32, 144)
    const float* __restrict__ b2,   // (32)
    const float* __restrict__ fcw,  // (4, 32)
    const float* __restrict__ fcb,  // (4)
    float* __restrict__ out,        // (B, 4)
    int nGroups)
{
    // Per-wave private LDS slices (no cross-wave sharing).
    __shared__ __align__(16) float smX[WAVES_PER_BLOCK][SAMPLES_PER_GROUP * 81];   // x tile, reused for h2
    __shared__ __align__(16) float smP2[WAVES_PER_BLOCK][SAMPLES_PER_GROUP * 144]; // conv2 activations

    const int wave = threadIdx.x >> 5;
    const int lane = threadIdx.x & 31;
    const int laneHi = (lane >> 4) & 1;   // 0: K+{0,1}, 1: K+{2,3} per WMMA f32 layout
    const int nlo = lane & 15;            // column index (channel) / A row (sample)

    int g = blockIdx.x * WAVES_PER_BLOCK + wave;
    if (g >= nGroups) g = nGroups - 1;    // duplicate work, same values: keeps barriers uniform

    // ---- Stage x tile (16 samples x 81 f32, contiguous) into LDS ----
    const float* xg = x + (size_t)g * (SAMPLES_PER_GROUP * 81);
    float* sx = smX[wave];
    for (int i = lane; i < SAMPLES_PER_GROUP * 81; i += 32) sx[i] = xg[i];
    __syncthreads();

    // ---- Conv1 B fragments from w1 (16x9), K padded to 12 with zeros ----
    // B[k][n] = w1[n][k]; lane holds n=nlo, k = 4kk + 2*laneHi + {0,1}
    v2f bf1[3];
#pragma unroll
    for (int kk = 0; kk < 3; ++kk) {
        int k0 = 4 * kk + 2 * laneHi;
        bf1[kk].x = (k0     <= 8) ? w1[nlo * 9 + k0]     : 0.0f;
        bf1[kk].y = (k0 + 1 <= 8) ? w1[nlo * 9 + k0 + 1] : 0.0f;
    }

    // ---- Conv1 A-offset precompute: patch k -> (di*9 + dj); invalid k clamped
    // (padded-K products are killed by the zeroed B fragment above) ----
    int aoff[3][2];
#pragma unroll
    for (int kk = 0; kk < 3; ++kk) {
#pragma unroll
        for (int t = 0; t < 2; ++t) {
            int k = 4 * kk + 2 * laneHi + t;
            aoff[kk][t] = (k <= 8) ? ((k / 3) * 9 + (k % 3)) : 0;
        }
    }

    // ---- Conv1 + maxpool (pool pre-activation; sigmoid is monotone, relu no-op) ----
    // C layout: lane owns channel nlo; c[r] = sample (laneHi ? r+8 : r).
    float pool[9][8];
#pragma unroll
    for (int c = 0; c < 9; ++c)
#pragma unroll
        for (int r = 0; r < 8; ++r) pool[c][r] = -3.4e38f;

    const float* sxs = sx + nlo * 81;  // A reads: lane L and L+16 both need sample nlo
#pragma unroll
    for (int oh = 0; oh < 6; ++oh) {
#pragma unroll
        for (int ow = 0; ow < 6; ++ow) {
            const int base = oh * 9 + ow;
            v8f c1 = {};
#pragma unroll
            for (int kk = 0; kk < 3; ++kk) {
                v2f a;
                a.x = sxs[base + aoff[kk][0]];
                a.y = sxs[base + aoff[kk][1]];
                c1 = __builtin_amdgcn_wmma_f32_16x16x4_f32(
                        false, a, false, bf1[kk], (short)0, c1, false, false);
            }
            const int cell = (oh >> 1) * 3 + (ow >> 1);
#pragma unroll
            for (int r = 0; r < 8; ++r) pool[cell][r] = fmaxf(pool[cell][r], c1[r]);
        }
    }

    // ---- Activation + write p2 = sigmoid(pool + b1) to LDS as [sample][c*9+cell] ----
    const float b1v = b1[nlo];
    float* sp = smP2[wave];
#pragma unroll
    for (int r = 0; r < 8; ++r) {
        const int s = laneHi ? (r + 8) : r;
#pragma unroll
        for (int cell = 0; cell < 9; ++cell) {
            sp[s * 144 + nlo * 9 + cell] = sigmoidf_fast(pool[cell][r] + b1v);
        }
    }
    __syncthreads();

    // ---- Conv2: (16 samples) x (32 ch) = p2(16x144) * w2^T, 36 K4 steps, 2 N tiles ----
    v8f c2a = {}, c2b = {};
    const float* sps = sp + nlo * 144;
    const int kbase = 2 * laneHi;
#pragma unroll 4
    for (int kk = 0; kk < 36; ++kk) {
        const int k0 = 4 * kk + kbase;
        v2f a  = *(const v2f*)(sps + k0);                    // 8B-aligned LDS b64
        v2f bA = *(const v2f*)(w2 + nlo * 144 + k0);         // B[k][n]=w2[n][k], tile 0
        v2f bB = *(const v2f*)(w2 + (16 + nlo) * 144 + k0);  // tile 1 (ch 16..31)
        c2a = __builtin_amdgcn_wmma_f32_16x16x4_f32(false, a, false, bA, (short)0, c2a, false, false);
        c2b = __builtin_amdgcn_wmma_f32_16x16x4_f32(false, a, false, bB, (short)0, c2b, false, false);
    }

    // ---- h2 = sigmoid(C2 + b2) staged to LDS (reuse x-tile region) as [sample][32] ----
    const float b2a = b2[nlo];
    const float b2b = b2[16 + nlo];
    __syncthreads();                       // conv1 x-tile reads fully done block-wide
    float* sh = smX[wave];
#pragma unroll
    for (int r = 0; r < 8; ++r) {
        const int s = laneHi ? (r + 8) : r;
        sh[s * 32 + nlo]      = sigmoidf_fast(c2a[r] + b2a);
        sh[s * 32 + 16 + nlo] = sigmoidf_fast(c2b[r] + b2b);
    }
    __syncthreads();

    // ---- FC: 16 samples x 4 outputs = 64 values, 2 per lane, coalesced stores ----
    float* og = out + (size_t)g * (SAMPLES_PER_GROUP * 4);
#pragma unroll
    for (int t = 0; t < 2; ++t) {
        const int idx = lane + 32 * t;     // 0..63 == s*4 + o
        const int s = idx >> 2;
        const int o = idx & 3;
        float acc = fcb[o];
#pragma unroll
        for (int n = 0; n < 32; ++n) acc += sh[s * 32 + n] * fcw[o * 32 + n];
        og[idx] = acc;
    }
}

extern "C" void kernel_launch(void* const* d_in, const int* in_sizes, int n_in,
                              void* d_out, int out_size, void* d_ws, size_t ws_size,
                              hipStream_t stream) {
    (void)n_in; (void)out_size; (void)d_ws; (void)ws_size;
    const float* x   = (const float*)d_in[0];
    const float* w1  = (const float*)d_in[1];
    const float* b1  = (const float*)d_in[2];
    const float* w2  = (const float*)d_in[3];
    const float* b2  = (const float*)d_in[4];
    const float* fcw = (const float*)d_in[5];
    const float* fcb = (const float*)d_in[6];
    float* out = (float*)d_out;

    const int nSamples = in_sizes[0] / 81;                       // 131072
    const int nGroups  = nSamples / SAMPLES_PER_GROUP;           // 8192
    const int nBlocks  = (nGroups + WAVES_PER_BLOCK - 1) / WAVES_PER_BLOCK;

    olcnn_kernel<<<dim3(nBlocks), dim3(128), 0, stream>>>(
        x, w1, b1, w2, b2, fcw, fcb, out, nGroups);
}